// MultiheadGEAttention_22016002359586
// MI455X (gfx1250) — compile-verified
//
#include <hip/hip_runtime.h>
#include <hip/hip_bf16.h>
#include <math.h>
#include <stdint.h>

// ---------------------------------------------------------------------------
// MultiheadGEAttention forward for MI455X (gfx1250), wave32, WMMA f32 path,
// with async global->LDS double-buffered staging in the projection GEMMs.
// Shapes: D=512, H=8, d_head=64, B=8, T=769 (N_KEEP+1), N_COLS=2048, D_COL=22.
// ---------------------------------------------------------------------------

#define D_MODEL 512
#define NHEADS  8
#define DHEAD   64
#define TSEQ    769
#define BATCH   8
#define NKEEP   768
#define NCOLS   2048
#define DCOL    22
#define DCP     24              // D_COL padded to multiple of 4 for K=4 WMMA
#define MROWS   (BATCH * TSEQ)  // 6152
#define QTILES  49              // ceil(769/16)
#define KGROUPS 13              // ceil(49/4) k-tiles per 4-wave block

typedef __attribute__((ext_vector_type(2))) float v2f;
typedef __attribute__((ext_vector_type(8))) float v8f;

static __device__ __forceinline__ v8f wmma_f32_k4(v2f a, v2f b, v8f c) {
  // D = A(16x4,f32) x B(4x16,f32) + C(16x16,f32); emits v_wmma_f32_16x16x4_f32
  return __builtin_amdgcn_wmma_f32_16x16x4_f32(
      /*neg_a=*/false, a, /*neg_b=*/false, b,
      /*c_mod=*/(short)0, c, /*reuse_a=*/false, /*reuse_b=*/false);
}

static __device__ __forceinline__ uint32_t lds_addr32(const void* p) {
  // Generic pointer to LDS: hardware LDS address is addr[31:0] (ISA 10.2).
  return (uint32_t)(uintptr_t)p;
}

static __device__ __forceinline__ void async_b128(uint32_t lds_off,
                                                  uint32_t voff,
                                                  const float* base) {
  // GVS mode: copies 16 bytes from SGPR base plus 32-bit VGPR byte offset to the LDS address in lds_off.
  asm volatile("global_load_async_to_lds_b128 %0, %1, %2"
               :
               : "v"(lds_off), "v"(voff), "s"(base)
               : "memory");
}

static __device__ __forceinline__ void wait_async0() {
  asm volatile("s_wait_asynccnt 0x0" ::: "memory");
}

// ---------------------------------------------------------------------------
// Y[M,512] = X[M,512] @ W[512,512]^T + bias, then Y *= smul * scale[col]
// (scale==nullptr -> 1). Used for f_head (scale=rel_emb, smul=1/8),
// f_tail, f_v, and the final output projection.
// Block: 128 threads (4 waves). Wave w owns the 16x16 tile at col0 + 16*w.
// Double-buffered async global->LDS staging (ASYNCcnt + barrier discipline).
// ---------------------------------------------------------------------------
__global__ void __launch_bounds__(128)
proj_gemm_kernel(const float* __restrict__ X,
                 const float* __restrict__ W,
                 const float* __restrict__ bias,
                 const float* __restrict__ scale,
                 float smul,
                 float* __restrict__ Y,
                 int M) {
  __shared__ float lds_a[2][16 * 64];   // X tile: 16 rows x 64 k
  __shared__ float lds_b[2][64 * 64];   // W tile: 64 out-cols x 64 k

  const int tid  = threadIdx.x;
  const int lane = tid & 31;
  const int wave = tid >> 5;
  const int row0 = blockIdx.x * 16;
  const int col0 = blockIdx.y * 64;
  const int n    = lane & 15;
  const int kh   = (lane >> 4) << 1;  // 0 (lanes 0-15) or 2 (lanes 16-31)
  const int col  = col0 + wave * 16 + n;

  // Per-thread staging geometry (A: 2x b128, B: 8x b128 per stage).
  const int ea = tid * 8;          // element index into the 16x64 A tile
  const int ra = ea >> 6;
  const int ka = ea & 63;
  int grow = row0 + ra;
  if (grow >= M) grow = M - 1;     // clamp: keeps EXEC full; masked at store
  const uint32_t a_voff0 = (uint32_t)((grow * D_MODEL + ka) * 4);

  const float bv = bias[col];
  v8f c;
#pragma unroll
  for (int g = 0; g < 8; ++g) c[g] = bv;

  // Issue one k-stage of async copies into buffer `buf`.
  auto issue = [&](int k0, int buf) {
    const uint32_t ldsa = lds_addr32(&lds_a[buf][ra * 64 + ka]);
    const uint32_t va   = a_voff0 + (uint32_t)(k0 * 4);
    async_b128(ldsa, va, X);
    async_b128(ldsa + 16u, va + 16u, X);
#pragma unroll
    for (int i = 0; i < 8; ++i) {
      const int e4 = tid + i * 128;
      const int rb = e4 >> 4;
      const int kb = (e4 & 15) << 2;
      const uint32_t ldsb = lds_addr32(&lds_b[buf][rb * 64 + kb]);
      const uint32_t vb   = (uint32_t)(((col0 + rb) * D_MODEL + k0 + kb) * 4);
      async_b128(ldsb, vb, W);
    }
  };

  issue(0, 0);

  for (int s = 0; s < 8; ++s) {
    const int buf = s & 1;
    wait_async0();       // our stage-s copies have landed in LDS
    __syncthreads();     // everyone's copies visible; prev compute finished
    if (s < 7) issue((s + 1) * 64, buf ^ 1);  // overlap with compute below

    const float* pa = &lds_a[buf][(lane & 15) * 64];
    const float* pb = &lds_b[buf][(wave * 16 + n) * 64];
#pragma unroll
    for (int kk = 0; kk < 64; kk += 4) {
      v2f a, b;
      a.x = pa[kk + kh];
      a.y = pa[kk + kh + 1];
      b.x = pb[kk + kh];
      b.y = pb[kk + kh + 1];
      c = wmma_f32_k4(a, b, c);
    }
  }

  const float sc = smul * (scale ? scale[col] : 1.0f);
#pragma unroll
  for (int g = 0; g < 8; ++g) {
    const int row = row0 + g + ((lane >> 4) << 3);
    if (row < M) Y[(size_t)row * D_MODEL + col] = c[g] * sc;
  }
}

// ---------------------------------------------------------------------------
// L2-normalize col_head/col_tail rows, gather per ids_keep (+CLS row 0),
// write padded [B*H*T, 24] buffers (last 2 lanes zero).
// ---------------------------------------------------------------------------
__global__ void __launch_bounds__(256)
gather_norm_kernel(const float* __restrict__ col_head,
                   const float* __restrict__ col_tail,
                   const int* __restrict__ ids_keep,
                   float* __restrict__ ch_g,
                   float* __restrict__ ct_g) {
  const int idx   = blockIdx.x * blockDim.x + threadIdx.x;
  const int nrows = BATCH * NHEADS * TSEQ;
  if (idx >= 2 * nrows) return;
  const int which = (idx >= nrows) ? 1 : 0;
  const int r     = idx - which * nrows;      // r = (b*H + h)*T + t
  const int t     = r % TSEQ;
  const int bh    = r / TSEQ;
  const int h     = bh % NHEADS;
  const int b     = bh / NHEADS;

  const float* src = which ? col_tail : col_head;
  float*       dst = which ? ct_g : ch_g;

  const int row = (t == 0) ? 0 : (ids_keep[b * NKEEP + (t - 1)] + 1);
  const float* p = &src[((size_t)h * NCOLS + row) * DCOL];

  float v[DCOL];
  float ss = 0.f;
#pragma unroll
  for (int j = 0; j < DCOL; ++j) {
    v[j] = p[j];
    ss += v[j] * v[j];
  }
  const float rn = rsqrtf(ss);
  float* q = &dst[(size_t)r * DCP];
#pragma unroll
  for (int j = 0; j < DCOL; ++j) q[j] = v[j] * rn;
  q[DCOL]     = 0.f;
  q[DCOL + 1] = 0.f;
}

// ---------------------------------------------------------------------------
// Fused score: logits[b,h,q,k] = fh_scaled[q]·ft[k]           (WMMA, K=64)
//            + mask from sigmoid(ch[q]·ct[k] + bias) > 0.5    (WMMA, K=24)
// fh is pre-scaled by rel_emb * (1/sqrt(64)) in proj_gemm_kernel.
// Block: 128 threads; waves take 4 consecutive k-tiles of one q-tile.
// ---------------------------------------------------------------------------
__global__ void __launch_bounds__(128)
score_kernel(const float* __restrict__ fh,
             const float* __restrict__ ft,
             const float* __restrict__ chg,
             const float* __restrict__ ctg,
             const float* __restrict__ bias1,
             float* __restrict__ logits) {
  __shared__ float lds_q[16 * 64];
  __shared__ float lds_k[4][16 * 64];
  __shared__ float lds_cq[16 * DCP];
  __shared__ float lds_ck[4][16 * DCP];

  const int tid  = threadIdx.x;
  const int lane = tid & 31;
  const int wave = tid >> 5;
  const int bh   = blockIdx.y;       // b*H + h
  const int b    = bh >> 3;
  const int h    = bh & 7;
  const int qt   = blockIdx.x / KGROUPS;
  const int kg   = blockIdx.x % KGROUPS;
  const int kt   = kg * 4 + wave;
  const int n    = lane & 15;
  const int kh   = (lane >> 4) << 1;

  // stage fh q-tile (16 x 64)
  {
    const int e  = tid * 8;
    const int r  = e >> 6;
    const int kk = e & 63;
    const int q  = qt * 16 + r;
    float4 z0 = make_float4(0.f, 0.f, 0.f, 0.f), z1 = z0;
    if (q < TSEQ) {
      const float4* src =
          (const float4*)&fh[((size_t)b * TSEQ + q) * D_MODEL + h * DHEAD + kk];
      z0 = src[0];
      z1 = src[1];
    }
    *(float4*)&lds_q[r * 64 + kk]     = z0;
    *(float4*)&lds_q[r * 64 + kk + 4] = z1;
  }
  // stage ft tiles for 4 waves (4 x 16 x 64)
  {
#pragma unroll
    for (int i = 0; i < 8; ++i) {
      const int e4 = tid + i * 128;       // float4 index, 1024 total
      const int w  = e4 >> 8;             // 256 float4 per tile
      const int r  = (e4 >> 4) & 15;
      const int kk = (e4 & 15) << 2;
      const int k  = (kg * 4 + w) * 16 + r;
      float4 z = make_float4(0.f, 0.f, 0.f, 0.f);
      if (k < TSEQ)
        z = *(const float4*)&ft[((size_t)b * TSEQ + k) * D_MODEL + h * DHEAD + kk];
      *(float4*)&lds_k[w][r * 64 + kk] = z;
    }
  }
  // stage normalized column vectors
  for (int e = tid; e < 16 * DCP; e += 128) {
    const int r = e / DCP, j = e % DCP;
    const int q = qt * 16 + r;
    lds_cq[e] = (q < TSEQ) ? chg[((size_t)bh * TSEQ + q) * DCP + j] : 0.f;
  }
  for (int e = tid; e < 4 * 16 * DCP; e += 128) {
    const int w = e / (16 * DCP);
    const int rem = e % (16 * DCP);
    const int r = rem / DCP, j = rem % DCP;
    const int k = (kg * 4 + w) * 16 + r;
    lds_ck[w][rem] = (k < TSEQ) ? ctg[((size_t)bh * TSEQ + k) * DCP + j] : 0.f;
  }
  __syncthreads();

  if (kt >= QTILES) return;  // wave-uniform exit (after barrier)

  v8f cw, cts;
#pragma unroll
  for (int g = 0; g < 8; ++g) { cw[g] = 0.f; cts[g] = 0.f; }

  const float* pq = &lds_q[(lane & 15) * 64];
  const float* pk = &lds_k[wave][(lane & 15) * 64];
#pragma unroll
  for (int kk = 0; kk < 64; kk += 4) {
    v2f a, bb;
    a.x  = pq[kk + kh];  a.y  = pq[kk + kh + 1];
    bb.x = pk[kk + kh];  bb.y = pk[kk + kh + 1];
    cw = wmma_f32_k4(a, bb, cw);
  }
  const float* pcq = &lds_cq[(lane & 15) * DCP];
  const float* pck = &lds_ck[wave][(lane & 15) * DCP];
#pragma unroll
  for (int kk = 0; kk < DCP; kk += 4) {
    v2f a, bb;
    a.x  = pcq[kk + kh]; a.y  = pcq[kk + kh + 1];
    bb.x = pck[kk + kh]; bb.y = pck[kk + kh + 1];
    cts = wmma_f32_k4(a, bb, cts);
  }

  const float bia = bias1[0];
#pragma unroll
  for (int g = 0; g < 8; ++g) {
    const int q = qt * 16 + g + ((lane >> 4) << 3);
    const int k = kt * 16 + (lane & 15);
    if (q < TSEQ && k < TSEQ) {
      float prob = 1.f / (1.f + __expf(-(cts[g] + bia)));
      if (q == k || k == 0) prob = 0.f;               // nsi + readout masks
      const float mask = (prob > 0.5f) ? 0.f : -10000.f;
      logits[((size_t)bh * TSEQ + q) * TSEQ + k] = cw[g] + mask;
    }
  }
}

// ---------------------------------------------------------------------------
// In-place row softmax over k (T=769) for fr_graph. One 256-thread block/row.
// ---------------------------------------------------------------------------
__global__ void __launch_bounds__(256)
softmax_kernel(float* __restrict__ fr) {
  __shared__ float red[256];
  const int row = blockIdx.x;
  const int tid = threadIdx.x;
  float* p = &fr[(size_t)row * TSEQ];

  float m = -3.4e38f;
  for (int k = tid; k < TSEQ; k += 256) m = fmaxf(m, p[k]);
  red[tid] = m;
  __syncthreads();
  for (int s = 128; s > 0; s >>= 1) {
    if (tid < s) red[tid] = fmaxf(red[tid], red[tid + s]);
    __syncthreads();
  }
  m = red[0];
  __syncthreads();

  float sum = 0.f;
  for (int k = tid; k < TSEQ; k += 256) sum += __expf(p[k] - m);
  red[tid] = sum;
  __syncthreads();
  for (int s = 128; s > 0; s >>= 1) {
    if (tid < s) red[tid] += red[tid + s];
    __syncthreads();
  }
  const float inv = 1.f / red[0];
  for (int k = tid; k < TSEQ; k += 256) p[k] = __expf(p[k] - m) * inv;
}

// ---------------------------------------------------------------------------
// out[b,q,h*64+d] = sum_k fr[b,h,q,k] * f_v[b,k,h*64+d]   (K = 769, WMMA)
// Block: 128 threads; each wave owns one of the 4 d-subtiles of head h.
// ---------------------------------------------------------------------------
__global__ void __launch_bounds__(128)
attn_kernel(const float* __restrict__ fr,
            const float* __restrict__ fv,
            float* __restrict__ out) {
  __shared__ float lds_a[16 * 64];   // fr tile: 16 q x 64 k
  __shared__ float lds_b[64 * 64];   // fv tile: 64 k x 64 d

  const int tid  = threadIdx.x;
  const int lane = tid & 31;
  const int wave = tid >> 5;
  const int bh   = blockIdx.y;
  const int b    = bh >> 3;
  const int h    = bh & 7;
  const int q0   = blockIdx.x * 16;
  const int n    = lane & 15;
  const int kh   = (lane >> 4) << 1;
  const int dn   = wave * 16 + n;

  v8f c;
#pragma unroll
  for (int g = 0; g < 8; ++g) c[g] = 0.f;

  for (int k0 = 0; k0 < 832; k0 += 64) {  // 13 blocks cover K=769 (zero pad)
    // stage fr tile (scalar: 769 not vector-aligned at the tail)
    {
      const int e  = tid * 8;
      const int r  = e >> 6;
      const int kk = e & 63;
      const int q  = q0 + r;
#pragma unroll
      for (int j = 0; j < 8; ++j) {
        const int k = k0 + kk + j;
        lds_a[r * 64 + kk + j] =
            (q < TSEQ && k < TSEQ) ? fr[((size_t)bh * TSEQ + q) * TSEQ + k] : 0.f;
      }
    }
    // stage fv tile
    {
#pragma unroll
      for (int i = 0; i < 8; ++i) {
        const int e4 = tid + i * 128;
        const int r  = e4 >> 4;
        const int kk = (e4 & 15) << 2;
        const int k  = k0 + r;
        float4 z = make_float4(0.f, 0.f, 0.f, 0.f);
        if (k < TSEQ)
          z = *(const float4*)&fv[((size_t)b * TSEQ + k) * D_MODEL + h * DHEAD + kk];
        *(float4*)&lds_b[r * 64 + kk] = z;
      }
    }
    __syncthreads();

    const float* pa = &lds_a[(lane & 15) * 64];
#pragma unroll
    for (int kk = 0; kk < 64; kk += 4) {
      v2f a, bb;
      a.x  = pa[kk + kh];
      a.y  = pa[kk + kh + 1];
      bb.x = lds_b[(kk + kh) * 64 + dn];
      bb.y = lds_b[(kk + kh + 1) * 64 + dn];
      c = wmma_f32_k4(a, bb, c);
    }
    __syncthreads();
  }

#pragma unroll
  for (int g = 0; g < 8; ++g) {
    const int q = q0 + g + ((lane >> 4) << 3);
    if (q < TSEQ)
      out[((size_t)b * TSEQ + q) * D_MODEL + h * DHEAD + dn] = c[g];
  }
}

// ---------------------------------------------------------------------------
extern "C" void kernel_launch(void* const* d_in, const int* in_sizes, int n_in,
                              void* d_out, int out_size, void* d_ws,
                              size_t ws_size, hipStream_t stream) {
  (void)in_sizes; (void)n_in; (void)out_size; (void)ws_size;

  const float* x_head   = (const float*)d_in[0];
  const float* x_tail   = (const float*)d_in[1];
  const int*   ids_keep = (const int*)d_in[2];
  const float* W_head_w = (const float*)d_in[3];
  const float* W_head_b = (const float*)d_in[4];
  const float* W_v_w    = (const float*)d_in[5];
  const float* W_v_b    = (const float*)d_in[6];
  const float* W_out_w  = (const float*)d_in[7];
  const float* W_out_b  = (const float*)d_in[8];
  const float* rel_emb  = (const float*)d_in[9];
  const float* col_head = (const float*)d_in[10];
  const float* col_tail = (const float*)d_in[11];
  const float* bias     = (const float*)d_in[12];

  float* ws     = (float*)d_ws;
  float* f_head = ws;
  float* f_tail = f_head + (size_t)MROWS * D_MODEL;
  float* f_v    = f_tail + (size_t)MROWS * D_MODEL;
  float* attn   = f_v    + (size_t)MROWS * D_MODEL;
  float* ch_g   = attn   + (size_t)MROWS * D_MODEL;
  float* ct_g   = ch_g   + (size_t)BATCH * NHEADS * TSEQ * DCP;

  float* x_out = (float*)d_out;
  float* fr    = x_out + (size_t)MROWS * D_MODEL;  // fr_graph output region

  const dim3 blk128(128);
  const dim3 gproj((MROWS + 15) / 16, D_MODEL / 64);

  // Projections: f_head gets rel_emb * (1/sqrt(d_head)) folded in.
  hipLaunchKernelGGL(proj_gemm_kernel, gproj, blk128, 0, stream,
                     x_head, W_head_w, W_head_b, rel_emb, 0.125f, f_head, MROWS);
  hipLaunchKernelGGL(proj_gemm_kernel, gproj, blk128, 0, stream,
                     x_tail, W_head_w, W_head_b, (const float*)nullptr, 1.0f,
                     f_tail, MROWS);
  hipLaunchKernelGGL(proj_gemm_kernel, gproj, blk128, 0, stream,
                     x_tail, W_v_w, W_v_b, (const float*)nullptr, 1.0f,
                     f_v, MROWS);

  // Normalize + gather column embeddings.
  const int nrows2 = 2 * BATCH * NHEADS * TSEQ;
  hipLaunchKernelGGL(gather_norm_kernel, dim3((nrows2 + 255) / 256), dim3(256),
                     0, stream, col_head, col_tail, ids_keep, ch_g, ct_g);

  // Fused score + adjacency mask -> logits in fr_graph region.
  hipLaunchKernelGGL(score_kernel, dim3(QTILES * KGROUPS, BATCH * NHEADS),
                     blk128, 0, stream, f_head, f_tail, ch_g, ct_g, bias, fr);

  // In-place softmax -> fr_graph (second output).
  hipLaunchKernelGGL(softmax_kernel, dim3(BATCH * NHEADS * TSEQ), dim3(256),
                     0, stream, fr);

  // Attention-weighted V, written un-transposed as [B*T, D].
  hipLaunchKernelGGL(attn_kernel, dim3(QTILES, BATCH * NHEADS), blk128, 0,
                     stream, fr, f_v, attn);

  // Output projection straight into d_out.
  hipLaunchKernelGGL(proj_gemm_kernel, gproj, blk128, 0, stream,
                     attn, W_out_w, W_out_b, (const float*)nullptr, 1.0f,
                     x_out, MROWS);
}